// HeteroMessagePassingLayer_57492432224468
// MI455X (gfx1250) — compile-verified
//
#include <hip/hip_runtime.h>

typedef float v2f __attribute__((ext_vector_type(2)));
typedef float v8f __attribute__((ext_vector_type(8)));

#define D 128
// XOR-swizzled LDS index: rows k+2,k+3 shift banks by 32 vs rows k,k+1 so the
// two half-wave groups (kb=0 vs kb=2) never collide on the same bank.
__device__ __forceinline__ int widx(int r, int c) {
    return r * D + (c ^ ((r & 2) << 4));
}

// ---------------------------------------------------------------------------
// Edge scatter: one wave per edge. Lane L moves floats [4L,4L+4) of the 512B
// source row with a coalesced b128 load + 4 coalesced global_atomic_add_f32.
// ---------------------------------------------------------------------------
__global__ __launch_bounds__(256) void scatter_rows(
    const float* __restrict__ x, const int* __restrict__ src,
    const int* __restrict__ dst, float* __restrict__ agg,
    float* __restrict__ cnt, int E)
{
    int gid  = blockIdx.x * 256 + threadIdx.x;
    int e    = gid >> 5;
    if (e >= E) return;
    int lane = gid & 31;
    int s = src[e];
    int d = dst[e];
    const float4 v = reinterpret_cast<const float4*>(x + (size_t)s * D)[lane];
    float* a = agg + (size_t)d * D + lane * 4;
    unsafeAtomicAdd(a + 0, v.x);
    unsafeAtomicAdd(a + 1, v.y);
    unsafeAtomicAdd(a + 2, v.z);
    unsafeAtomicAdd(a + 3, v.w);
    if (lane == 0) unsafeAtomicAdd(cnt + d, 1.0f);
}

// cnt <- 1 / max(cnt, 1)
__global__ __launch_bounds__(256) void reciprocal_cnt(float* __restrict__ c, int N)
{
    int i = blockIdx.x * 256 + threadIdx.x;
    if (i < N) c[i] = 1.0f / fmaxf(c[i], 1.0f);
}

// ---------------------------------------------------------------------------
// Fused node GEMM:  out[m, :] = X[m,:]@Ws^T + b + (Agg[m,:]*inv[m])@Wr^T
// 8 waves/block, each wave computes a 16x128 output tile via
// V_WMMA_F32_16X16X4_F32 (full fp32 precision). Weights staged transposed in
// LDS (64KB, reused for both passes).
//
// V_WMMA_F32_16X16X4_F32 operand layout (wave32):
//   A (16x4 MxK): lane&15 = M;  VGPR0/1 = K = (lane>>4)*2 + {0,1}
//   B (4x16 KxN): lane&15 = N;  VGPR0/1 = K = (lane>>4)*2 + {0,1}
//   C/D (16x16):  lane&15 = N;  VGPRv   = M = v + 8*(lane>>4)
// ---------------------------------------------------------------------------
__global__ __launch_bounds__(256) void gemm_fused(
    const float* __restrict__ X, const float* __restrict__ Agg,
    const float* __restrict__ invc,
    const float* __restrict__ Ws, const float* __restrict__ bs,
    const float* __restrict__ Wr, float* __restrict__ out, int N)
{
    __shared__ float wT[D * D];   // 64 KB: transposed weight, swizzled

    const int tid  = threadIdx.x;
    const int lane = tid & 31;
    const int wave = tid >> 5;
    const int m0   = blockIdx.x * 128 + wave * 16;
    const int mrow = lane & 15;          // M for A, N for B/C/D
    const int kb   = (lane >> 4) << 1;   // 0 or 2
    const bool active = (m0 < N);        // N % 16 == 0, so full tiles

    // stage Ws^T: wT[i][o] = Ws[o][i]; consecutive tid -> consecutive o
    // (conflict-free LDS writes; strided global reads served by L0/L2)
    for (int e = tid; e < D * D; e += 256) {
        int i = e >> 7, o = e & 127;
        wT[widx(i, o)] = Ws[o * D + i];
    }
    __syncthreads();

    v8f acc[8];
    const float* xrow = X + (size_t)(m0 + mrow) * D;
    if (active) {
#pragma unroll
        for (int nt = 0; nt < 8; ++nt) {
            float bv = bs[nt * 16 + mrow];
            acc[nt] = (v8f){bv, bv, bv, bv, bv, bv, bv, bv};
        }
        for (int k = 0; k < D; k += 4) {
            v2f a = *reinterpret_cast<const v2f*>(xrow + k + kb);
#pragma unroll
            for (int nt = 0; nt < 8; ++nt) {
                v2f b;
                b.x = wT[widx(k + kb,     nt * 16 + mrow)];
                b.y = wT[widx(k + kb + 1, nt * 16 + mrow)];
                acc[nt] = __builtin_amdgcn_wmma_f32_16x16x4_f32(
                    false, a, false, b, (short)0, acc[nt], false, false);
            }
        }
    }
    __syncthreads();

    // stage Wr^T into the same LDS buffer
    for (int e = tid; e < D * D; e += 256) {
        int i = e >> 7, o = e & 127;
        wT[widx(i, o)] = Wr[o * D + i];
    }
    __syncthreads();
    if (!active) return;

    const float* arow = Agg + (size_t)(m0 + mrow) * D;
    const float  inv  = invc[m0 + mrow];
    for (int k = 0; k < D; k += 4) {
        v2f a = *reinterpret_cast<const v2f*>(arow + k + kb);
        a.x *= inv; a.y *= inv;
#pragma unroll
        for (int nt = 0; nt < 8; ++nt) {
            v2f b;
            b.x = wT[widx(k + kb,     nt * 16 + mrow)];
            b.y = wT[widx(k + kb + 1, nt * 16 + mrow)];
            acc[nt] = __builtin_amdgcn_wmma_f32_16x16x4_f32(
                false, a, false, b, (short)0, acc[nt], false, false);
        }
    }

    const int rbase = (lane >> 4) << 3;   // 0 or 8
#pragma unroll
    for (int nt = 0; nt < 8; ++nt) {
#pragma unroll
        for (int v = 0; v < 8; ++v) {
            out[(size_t)(m0 + rbase + v) * D + nt * 16 + mrow] = acc[nt][v];
        }
    }
}

// ---------------------------------------------------------------------------
extern "C" void kernel_launch(void* const* d_in, const int* in_sizes, int n_in,
                              void* d_out, int out_size, void* d_ws, size_t ws_size,
                              hipStream_t stream)
{
    const float* x_user      = (const float*)d_in[0];
    const float* x_item      = (const float*)d_in[1];
    const int*   ei_u2i      = (const int*)d_in[2];
    const int*   ei_i2u      = (const int*)d_in[3];
    const float* W_self_user = (const float*)d_in[4];
    const float* b_self_user = (const float*)d_in[5];
    const float* W_self_item = (const float*)d_in[6];
    const float* b_self_item = (const float*)d_in[7];
    const float* W_rates     = (const float*)d_in[8];
    const float* W_ratedby   = (const float*)d_in[9];

    const int n_user = in_sizes[0] / D;
    const int n_item = in_sizes[1] / D;
    const int Eu     = in_sizes[2] / 2;   // ei[0]=src users, ei[1]=dst items
    const int Ei     = in_sizes[3] / 2;

    // workspace layout (floats): agg_user | agg_item | cnt_user | cnt_item
    float* agg_user = (float*)d_ws;
    float* agg_item = agg_user + (size_t)n_user * D;
    float* cnt_user = agg_item + (size_t)n_item * D;
    float* cnt_item = cnt_user + n_user;
    const size_t zbytes = ((size_t)(n_user + n_item) * (D + 1)) * sizeof(float);
    hipMemsetAsync(d_ws, 0, zbytes, stream);

    dim3 blk(256);
    // (user, rates, item): user -> item
    scatter_rows<<<(unsigned)((Eu * 32 + 255) / 256), blk, 0, stream>>>(
        x_user, ei_u2i, ei_u2i + Eu, agg_item, cnt_item, Eu);
    // (item, rated_by, user): item -> user
    scatter_rows<<<(unsigned)((Ei * 32 + 255) / 256), blk, 0, stream>>>(
        x_item, ei_i2u, ei_i2u + Ei, agg_user, cnt_user, Ei);

    reciprocal_cnt<<<(n_user + 255) / 256, blk, 0, stream>>>(cnt_user, n_user);
    reciprocal_cnt<<<(n_item + 255) / 256, blk, 0, stream>>>(cnt_item, n_item);

    float* out_user = (float*)d_out;
    float* out_item = out_user + (size_t)n_user * D;

    gemm_fused<<<(n_user + 127) / 128, blk, 0, stream>>>(
        x_user, agg_user, cnt_user, W_self_user, b_self_user, W_ratedby,
        out_user, n_user);
    gemm_fused<<<(n_item + 127) / 128, blk, 0, stream>>>(
        x_item, agg_item, cnt_item, W_self_item, b_self_item, W_rates,
        out_item, n_item);
}